// OCRNet_10247791969021
// MI455X (gfx1250) — compile-verified
//
#include <hip/hip_runtime.h>
#include <hip/hip_bf16.h>
#include <stdint.h>
#include <math.h>

// ---------------- problem constants ----------------
#define Bn   4
#define Hh   64
#define Ww   64
#define Cc   128
#define NBOX (Hh*Ww)        // 4096
#define MAXB 32
#define MAXL 64
#define ROIH 8
#define SCALE_F 8.0f
#define IOU_T   0.5f
#define SCORE_T 0.3f

// ---------------- workspace layout (float elems) ----------------
#define WS_SCORES 0
#define WS_BOXES  (Bn*NBOX)                 // 16384
#define WS_FB     (WS_BOXES + Bn*NBOX*4)    // 81920  (+512)

// ---------------- output layout (elems) ----------------
#define O_NMS  0
#define O_HROI 512
#define O_W    (O_HROI + (size_t)Bn*MAXB*ROIH*MAXL*Cc)     // 8389120
#define O_VROI (O_W + Bn*MAXB)                             // 8389248
#define O_HT   (O_VROI + (size_t)Bn*MAXB*MAXL*ROIH*Cc)     // 16777856

typedef float v2f  __attribute__((ext_vector_type(2)));
typedef float v8f  __attribute__((ext_vector_type(8)));
typedef unsigned int u32x4 __attribute__((ext_vector_type(4)));
typedef int   i32x4 __attribute__((ext_vector_type(4)));
typedef int   i32x8 __attribute__((ext_vector_type(8)));

#ifndef __has_builtin
#define __has_builtin(x) 0
#endif
#if __has_builtin(__builtin_amdgcn_tensor_load_to_lds)
#define HAVE_TDM 1
#else
#define HAVE_TDM 0
#endif

// =====================================================================
// Kernel 1: scores = sigmoid(ch0); boxes = SCALE * [x-d1, y-d2, x+d3, y+d4]
// =====================================================================
__global__ void k_prep(const float* __restrict__ bbox, float* __restrict__ ws) {
  int idx = blockIdx.x * blockDim.x + threadIdx.x;
  if (idx >= Bn * NBOX) return;
  int rem = idx & (NBOX - 1);
  float xx = (float)(rem & (Ww - 1));
  float yy = (float)(rem >> 6);
  const float* p = bbox + (size_t)idx * 5;
  ws[WS_SCORES + idx] = 1.0f / (1.0f + expf(-p[0]));
  float* bx = ws + WS_BOXES + (size_t)idx * 4;
  bx[0] = SCALE_F * (xx - p[1]);
  bx[1] = SCALE_F * (yy - p[2]);
  bx[2] = SCALE_F * (xx + p[3]);
  bx[3] = SCALE_F * (yy + p[4]);
}

// =====================================================================
// Kernel 2: greedy NMS, one block (256 threads) per batch image.
// Scores cached in registers; validity = 4096-bit LDS mask; LDS argmax tree.
// =====================================================================
__global__ void __launch_bounds__(256) k_nms(float* __restrict__ ws,
                                             float* __restrict__ out) {
  __shared__ float rv[256];
  __shared__ int   ri[256];
  __shared__ unsigned vm[NBOX / 32];
  __shared__ float sb[4];
  __shared__ float sareaj;
  __shared__ int   ssel;
  __shared__ int   selA[MAXB];

  const int t = threadIdx.x, b = blockIdx.x;
  const float* scores = ws + WS_SCORES + (size_t)b * NBOX;
  const float* boxes  = ws + WS_BOXES  + (size_t)b * NBOX * 4;

  if (t < NBOX / 32) vm[t] = 0xFFFFFFFFu;
  float sc[16];
#pragma unroll
  for (int i = 0; i < 16; ++i) sc[i] = scores[t + i * 256];
  __syncthreads();

  for (int it = 0; it < MAXB; ++it) {
    // masked argmax (first-index tie-break to match jnp.argmax)
    float bv = -INFINITY; int bi = -1;
#pragma unroll
    for (int i = 0; i < 16; ++i) {
      int idx = t + i * 256;
      bool val = (vm[idx >> 5] >> (idx & 31)) & 1u;
      float s = val ? sc[i] : -INFINITY;
      if (s > bv || (s == bv && (unsigned)idx < (unsigned)bi)) { bv = s; bi = idx; }
    }
    rv[t] = bv; ri[t] = bi;
    __syncthreads();
    for (int off = 128; off > 0; off >>= 1) {
      if (t < off) {
        float ov = rv[t + off]; int oi = ri[t + off];
        if (ov > rv[t] || (ov == rv[t] && (unsigned)oi < (unsigned)ri[t])) {
          rv[t] = ov; ri[t] = oi;
        }
      }
      __syncthreads();
    }
    if (t == 0) {
      int j = ri[0];
      bool ok = rv[0] > -INFINITY;
      ssel = ok ? j : -1;
      selA[it] = ssel;
      if (ok) {
        const float* bj = boxes + (size_t)j * 4;
        sb[0] = bj[0]; sb[1] = bj[1]; sb[2] = bj[2]; sb[3] = bj[3];
        sareaj = (sb[2] - sb[0]) * (sb[3] - sb[1]);
      }
    }
    __syncthreads();
    int j = ssel;
    if (j >= 0) {
      float b0 = sb[0], b1 = sb[1], b2 = sb[2], b3 = sb[3], aj = sareaj;
#pragma unroll
      for (int i = 0; i < 16; ++i) {
        int idx = t + i * 256;
        if ((vm[idx >> 5] >> (idx & 31)) & 1u) {
          const float* bp = boxes + (size_t)idx * 4;
          float x1 = bp[0], y1 = bp[1], x2 = bp[2], y2 = bp[3];
          float iw = fmaxf(fminf(x2, b2) - fmaxf(x1, b0), 0.f);
          float ih = fmaxf(fminf(y2, b3) - fmaxf(y1, b1), 0.f);
          float inter = iw * ih;
          float area = (x2 - x1) * (y2 - y1);
          float iou = inter / (area + aj - inter + 1e-9f);
          if (iou > IOU_T) atomicAnd(&vm[idx >> 5], ~(1u << (idx & 31)));
        }
      }
    }
    __syncthreads();
  }

  if (t < MAXB) {
    int j = selA[t];
    bool ok = j >= 0;
    int jj = ok ? j : 0;
    const float* bj = boxes + (size_t)jj * 4;
    float s = scores[jj];
    bool keep = ok && (s >= SCORE_T);
    float o0 = keep ? bj[0] : 0.f, o1 = keep ? bj[1] : 0.f;
    float o2 = keep ? bj[2] : 0.f, o3 = keep ? bj[3] : 0.f;
    float* onms = out + O_NMS + ((size_t)b * MAXB + t) * 4;
    onms[0] = o0; onms[1] = o1; onms[2] = o2; onms[3] = o3;
    float* fb = ws + WS_FB + ((size_t)b * MAXB + t) * 4;
    fb[0] = o0 / SCALE_F; fb[1] = o1 / SCALE_F;
    fb[2] = o2 / SCALE_F; fb[3] = o3 / SCALE_F;
  }
}

// =====================================================================
// TDM helper: DMA a (t1 lines x t0 elems, stride s0 elems) f32 tile
// from global into LDS via the Tensor Data Mover.
// (6-arg builtin form: g0, g1, g2, g3, extra group, cpol)
// =====================================================================
#if HAVE_TDM
__device__ __forceinline__ void tdm_load_tile(uint32_t lds_off, const float* g,
                                              uint32_t t0, uint32_t t1,
                                              uint32_t s0) {
  uint64_t ga = (uint64_t)(uintptr_t)g;
  uint32_t ga_lo = (uint32_t)__builtin_amdgcn_readfirstlane((int)(uint32_t)ga);
  uint32_t ga_hi = (uint32_t)__builtin_amdgcn_readfirstlane((int)(uint32_t)(ga >> 32));
  t0      = (uint32_t)__builtin_amdgcn_readfirstlane((int)t0);
  t1      = (uint32_t)__builtin_amdgcn_readfirstlane((int)t1);
  s0      = (uint32_t)__builtin_amdgcn_readfirstlane((int)s0);
  lds_off = (uint32_t)__builtin_amdgcn_readfirstlane((int)lds_off);

  u32x4 g0;
  g0[0] = 1u;                                   // count=1 (valid), user mode
  g0[1] = lds_off;                              // D#.lds_addr (bytes)
  g0[2] = ga_lo;                                // global_addr[31:0]
  g0[3] = (ga_hi & 0x01FFFFFFu) | (2u << 30);   // global_addr[56:32], type=2

  i32x8 g1;
  g1[0] = (int)(2u << 16);                                      // data_size=2 (4B)
  g1[1] = (int)((t0 & 0xFFFFu) << 16);                          // tensor_dim0 lo16
  g1[2] = (int)(((t0 >> 16) & 0xFFFFu) | ((t1 & 0xFFFFu) << 16)); // td0 hi / td1 lo
  g1[3] = (int)(((t1 >> 16) & 0xFFFFu) | ((t0 & 0xFFFFu) << 16)); // td1 hi / tile_dim0
  g1[4] = (int)(t1 & 0xFFFFu);                                  // tile_dim1, tile_dim2=0
  g1[5] = (int)s0;                                              // tensor_dim0_stride lo32
  g1[6] = 0;
  g1[7] = 0;
  i32x4 z4 = {0, 0, 0, 0};
  i32x8 z8 = {0, 0, 0, 0, 0, 0, 0, 0};
  __builtin_amdgcn_tensor_load_to_lds(g0, g1, z4, z4, z8, 0);
}
#endif

__device__ __forceinline__ void wait_tensor0() {
#if __has_builtin(__builtin_amdgcn_s_wait_tensorcnt)
  __builtin_amdgcn_s_wait_tensorcnt(0);
#elif HAVE_TDM
  asm volatile("s_wait_tensorcnt 0" ::: "memory");
#endif
}

// =====================================================================
// Kernel 3: ROI pooling as WMMA GEMMs.
// One block per (batch, box, orientation). For each of the 8 fixed-index
// rows: out(64x128) = W0(64x64)@R0(64x128) + W1(64x64)@R1(64x128),
// where R0/R1 are fmap rows (H) or columns (V) staged in LDS by the TDM,
// and W0/W1 carry the reference's exact 4-tap weights (2 nnz per row).
// 8 waves: wave w owns channel tile nt=w; 4 M-tiles accumulated in VGPRs.
// =====================================================================
__global__ void __launch_bounds__(256) k_roi(const float* __restrict__ fmap,
                                             const float* __restrict__ ws,
                                             float* __restrict__ out) {
  __shared__ __align__(16) float R[2][MAXL][Cc];   // 64 KB

  const int tid = threadIdx.x;
  const int lane = tid & 31;
  const int wv = tid >> 5;          // wave id = channel tile nt (0..7)
  const int hi = (lane >> 4) & 1;   // K/M-half select per WMMA layout
  const int nl = lane & 15;

  const int blk   = blockIdx.x;     // 0..255
  const int horiz = ((blk & 1) == 0);
  const int box   = (blk >> 1) & 31;
  const int b     = blk >> 6;

  // ---- box geometry (uniform) ----
  const float* fbp = ws + WS_FB + ((size_t)b * MAXB + box) * 4;
  float f0 = fbp[0], f1 = fbp[1], f2 = fbp[2], f3 = fbp[3];
  float w = f2 - f0, h = f3 - f1;
  bool is_dir = horiz ? (w >= h) : (w < h);
  float bz0 = is_dir ? f0 : 0.f, bz1 = is_dir ? f1 : 0.f;
  float bz2 = is_dir ? f2 : 0.f, bz3 = is_dir ? f3 : 0.f;
  w = bz2 - bz0; h = bz3 - bz1;
  bool nz = (w >= 0.1f) || (h >= 0.1f);
  float bw = (w <= 0.f) ? 1e-4f : w;
  float bh = (h <= 0.f) ? 1e-4f : h;
  float ratio = nz ? (horiz ? bw / bh : bh / bw) : 0.f;
  int nlen = (int)fminf(ceilf(ratio * (float)ROIH), (float)MAXL);
  float lenf = fmaxf((float)nlen, 2.0f);
  float map_w = horiz ? bw / (lenf - 1.f) : bw / (float)(ROIH - 1);
  float map_h = horiz ? bh / (float)(ROIH - 1) : bh / (lenf - 1.f);

  if (tid == 0) {
    int* iout = (int*)out;
    if (horiz) iout[O_W  + b * MAXB + box] = nlen;
    else       iout[O_HT + b * MAXB + box] = nlen;
  }

  const float* img = fmap + (size_t)b * Hh * Ww * Cc;

  for (int s = 0; s < ROIH; ++s) {
    // ---- per-s uniform geometry: the "fixed" axis ----
    const float* src0; const float* src1;
    float sw0, sw1;               // H: p,q   V: u,v
    uint32_t t0e, t1e;
    if (horiz) {
      float ys = (float)s * map_h + bz1;
      int fy = (int)floorf(ys);
      int y0 = min(max(fy, 0), Hh - 1);
      int y1 = min(max(fy + 1, 0), Hh - 1);
      sw0 = (float)y1 - ys; sw1 = ys - (float)y0;
      src0 = img + (size_t)y0 * Ww * Cc;
      src1 = img + (size_t)y1 * Ww * Cc;
      t0e = Ww * Cc; t1e = 1;          // one contiguous 32 KB line
    } else {
      float xs = (float)s * map_w + bz0;
      int fx = (int)floorf(xs);
      int x0 = min(max(fx, 0), Ww - 1);
      int x1 = min(max(fx + 1, 0), Ww - 1);
      sw0 = (float)x1 - xs; sw1 = xs - (float)x0;
      src0 = img + (size_t)x0 * Cc;
      src1 = img + (size_t)x1 * Cc;
      t0e = Cc; t1e = Hh;              // 64 lines x 512 B, stride 32 KB
    }

    __syncthreads();                    // protect LDS WAR vs previous iter
#if HAVE_TDM
    if (tid < 32) {                     // wave 0 drives the TDM
      tdm_load_tile((uint32_t)(uintptr_t)&R[0][0][0], src0, t0e, t1e, Ww * Cc);
      tdm_load_tile((uint32_t)(uintptr_t)&R[1][0][0], src1, t0e, t1e, Ww * Cc);
      wait_tensor0();
    }
    __syncthreads();
#else
    {
      int w4 = (int)(t0e >> 2);
      int n4 = w4 * (int)t1e;           // 2048 float4 per tile
      float4* d0 = (float4*)&R[0][0][0];
      float4* d1 = (float4*)&R[1][0][0];
      for (int e = tid; e < n4; e += 256) {
        int row = e / w4, c4 = e - row * w4;
        d0[e] = *(const float4*)(src0 + (size_t)row * Ww * Cc + c4 * 4);
        d1[e] = *(const float4*)(src1 + (size_t)row * Ww * Cc + c4 * 4);
      }
      __syncthreads();
    }
#endif

    // ---- per-lane weight-matrix geometry: the "moving" axis (M rows) ----
    int k0a[4], k1a[4];
    float c00a[4], c01a[4], c10a[4], c11a[4];
#pragma unroll
    for (int mt = 0; mt < 4; ++mt) {
      int m = mt * 16 + nl;             // A-matrix row for this lane
      if (horiz) {
        float xm = (float)m * map_w + bz0;
        int fx = (int)floorf(xm);
        int x0 = min(max(fx, 0), Ww - 1);
        int x1 = min(max(fx + 1, 0), Ww - 1);
        float u = (float)x1 - xm, vv = xm - (float)x0;
        k0a[mt] = x0; k1a[mt] = x1;
        c00a[mt] = u * sw0;  c01a[mt] = u * sw1;   // wa, wb  (row y0)
        c10a[mt] = vv * sw0; c11a[mt] = vv * sw1;  // wc, wd  (row y1)
      } else {
        float ym = (float)m * map_h + bz1;
        int fy = (int)floorf(ym);
        int y0 = min(max(fy, 0), Hh - 1);
        int y1 = min(max(fy + 1, 0), Hh - 1);
        float p = (float)y1 - ym, q = ym - (float)y0;
        k0a[mt] = y0; k1a[mt] = y1;
        c00a[mt] = sw0 * p; c01a[mt] = sw1 * p;    // wa, wc  (col x0)
        c10a[mt] = sw0 * q; c11a[mt] = sw1 * q;    // wb, wd  (col x1)
      }
    }

    // ---- GEMM: acc += W0@R0 + W1@R1 via v_wmma_f32_16x16x4_f32 ----
    v8f acc[4];
#pragma unroll
    for (int mt = 0; mt < 4; ++mt) acc[mt] = (v8f){0, 0, 0, 0, 0, 0, 0, 0};

    for (int r = 0; r < 2; ++r) {
      const float(*Rr)[Cc] = R[r];
      float cr0[4], cr1[4];
#pragma unroll
      for (int mt = 0; mt < 4; ++mt) {
        cr0[mt] = r ? c10a[mt] : c00a[mt];
        cr1[mt] = r ? c11a[mt] : c01a[mt];
      }
#pragma unroll 4
      for (int ks = 0; ks < 16; ++ks) {
        int kb = ks * 4 + 2 * hi;
        v2f Bv = { Rr[kb][wv * 16 + nl], Rr[kb + 1][wv * 16 + nl] };
#pragma unroll
        for (int mt = 0; mt < 4; ++mt) {
          float a0 = ((kb     == k0a[mt]) ? cr0[mt] : 0.f) +
                     ((kb     == k1a[mt]) ? cr1[mt] : 0.f);
          float a1 = ((kb + 1 == k0a[mt]) ? cr0[mt] : 0.f) +
                     ((kb + 1 == k1a[mt]) ? cr1[mt] : 0.f);
          v2f Av = { a0, a1 };
          acc[mt] = __builtin_amdgcn_wmma_f32_16x16x4_f32(
              false, Av, false, Bv, (short)0, acc[mt], false, false);
        }
      }
    }

    // ---- masked coalesced stores (C/D layout: VGPR v -> M=v / v+8) ----
#pragma unroll
    for (int mt = 0; mt < 4; ++mt) {
#pragma unroll
      for (int v = 0; v < 8; ++v) {
        int ml = mt * 16 + v + hi * 8;
        float msk = (nz && (ml < nlen)) ? 1.f : 0.f;
        size_t off;
        if (horiz)
          off = O_HROI + ((((size_t)(b * MAXB + box) * ROIH + s) * MAXL + ml) * Cc
                          + wv * 16 + nl);
        else
          off = O_VROI + ((((size_t)(b * MAXB + box) * MAXL + ml) * ROIH + s) * Cc
                          + wv * 16 + nl);
        out[off] = acc[mt][v] * msk;
      }
    }
  }
}

// =====================================================================
extern "C" void kernel_launch(void* const* d_in, const int* in_sizes, int n_in,
                              void* d_out, int out_size, void* d_ws,
                              size_t ws_size, hipStream_t stream) {
  const float* fmap = (const float*)d_in[0];
  const float* bbox = (const float*)d_in[1];
  float* ws  = (float*)d_ws;
  float* out = (float*)d_out;

  k_prep<<<(Bn * NBOX + 255) / 256, 256, 0, stream>>>(bbox, ws);
  k_nms<<<Bn, 256, 0, stream>>>(ws, out);
  k_roi<<<Bn * MAXB * 2, 256, 0, stream>>>(fmap, ws, out);
}